// WbQuadratic_50113678410588
// MI455X (gfx1250) — compile-verified
//
#include <hip/hip_runtime.h>
#include <hip/hip_bf16.h>

// WB quadratic color correction:
//   out[n,c,h,w] = clip( sum_k feats[n,h,w,k] * (params[n,c*10+k]*10-5), 0, 1 )
//   feats = [b*b, g*g, r*r, b*g, b*r, g*r, b, g, r, 1]   (b,g,r = img channels 0,1,2)
//
// Memory-bound (~2.6 FLOP/B, 201 MB total traffic -> ~8.6us floor at 23.3 TB/s).
// Matvec mapped onto V_WMMA_F32_16X16X4_F32 (A = 16x12 padded coef, B = 12x16
// padded per-pixel features, 3 WMMAs accumulate K=12). Streaming uses NT
// load/store hints; half-wave data movement via ds_bpermute (__shfl_xor).

typedef __attribute__((ext_vector_type(2))) float v2f;
typedef __attribute__((ext_vector_type(8))) float v8f;

namespace {
constexpr int HW            = 512 * 512;   // pixels per plane
constexpr int NIMG          = 32;
constexpr int PIX_PER_ITER  = 32;          // two 16-pixel WMMA column sets
constexpr int ITERS         = 8;           // per wave
constexpr int PIX_PER_WAVE  = PIX_PER_ITER * ITERS;        // 256
constexpr int WAVES_PER_IMG = HW / PIX_PER_WAVE;           // 1024
constexpr int TOTAL_WAVES   = NIMG * WAVES_PER_IMG;        // 32768
constexpr int BLOCK         = 256;                         // 8 waves
constexpr int GRID          = TOTAL_WAVES / (BLOCK / 32);  // 4096
}

__global__ __launch_bounds__(BLOCK)
void WbQuadratic_wmma_kernel(const float* __restrict__ img,
                             const float* __restrict__ params,
                             float* __restrict__ out)
{
    const int lane = threadIdx.x & 31;
    const int wave = (blockIdx.x * BLOCK + threadIdx.x) >> 5;
    const int n    = wave / WAVES_PER_IMG;           // image id (wave-uniform)
    const int ww   = wave % WAVES_PER_IMG;
    const int pix0 = ww * PIX_PER_WAVE;

    const int  m  = lane & 15;                       // A-matrix row (channel)
    const bool hi = (lane & 16) != 0;                // upper half-wave
    const int  kh = hi ? 2 : 0;                      // K sub-offset per ISA layout

    // ---- Build A (coefficients), 16x4 f32 per K-block; rows>=3 / K>=10 are 0.
    // ISA 16x4 f32 A layout: VGPR0 = {K=kb+0 | K=kb+2}, VGPR1 = {K=kb+1 | K=kb+3}
    const float* pp = params + n * 30;
    const bool vm = (m < 3);
    auto coef = [&](int k, bool valid) -> float {
        int  idx = valid ? (m * 10 + k) : 0;         // branch-free (EXEC stays full)
        float t  = pp[idx];
        return valid ? (t * 10.0f - 5.0f) : 0.0f;
    };
    v2f A0, A1, A2;
    A0.x = coef(0 + kh, vm);            A0.y = coef(1 + kh, vm);
    A1.x = coef(4 + kh, vm);            A1.y = coef(5 + kh, vm);
    A2.x = coef(8 + kh, vm && !hi);     A2.y = coef(9 + kh, vm && !hi); // K=10,11 -> 0

    const float* bp = img + (size_t)n * 3 * HW;      // channel 0 plane ("b")
    const float* gp = bp + HW;
    const float* rp = gp + HW;
    float* ob = out + (size_t)n * 3 * HW;
    float* og = ob + HW;
    float* orp = og + HW;

    // One 16-pixel column set: builds B (12x16 features, rows per half-wave per
    // ISA layout) and accumulates 3x K=4 WMMAs.
    auto run16 = [&](float b, float g, float r) -> v8f {
        float bb = b * b, gg = g * g, rr = r * r;
        float bg = b * g, br = b * r, gr = g * r;
        v2f B0, B1, B2;
        B0.x = hi ? rr   : bb;   B0.y = hi ? bg   : gg;   // rows {0,1}/{2,3}
        B1.x = hi ? b    : br;   B1.y = hi ? g    : gr;   // rows {4,5}/{6,7}
        B2.x = hi ? 0.0f : r;    B2.y = hi ? 0.0f : 1.0f; // rows {8,9}/{10,11}
        v8f c = {};
        c = __builtin_amdgcn_wmma_f32_16x16x4_f32(false, A0, false, B0,
                                                  (short)0, c, false, false);
        c = __builtin_amdgcn_wmma_f32_16x16x4_f32(false, A1, false, B1,
                                                  (short)0, c, false, false);
        c = __builtin_amdgcn_wmma_f32_16x16x4_f32(false, A2, false, B2,
                                                  (short)0, c, false, false);
        return c;
    };

    #pragma unroll 2
    for (int it = 0; it < ITERS; ++it) {
        const int idx = pix0 + it * PIX_PER_ITER + lane;   // my pixel

        // Streamed, fully-coalesced NT loads (each byte read exactly once).
        float bm = __builtin_nontemporal_load(bp + idx);
        float gm = __builtin_nontemporal_load(gp + idx);
        float rm = __builtin_nontemporal_load(rp + idx);

        // Half-wave swap so both halves see the same 16 pixels (B layout needs
        // pixel j in lanes j and j+16).
        float bo = __shfl_xor(bm, 16, 32);
        float go = __shfl_xor(gm, 16, 32);
        float ro = __shfl_xor(rm, 16, 32);

        // Set X -> pixels [base, base+16), set Y -> [base+16, base+32)
        float bX = hi ? bo : bm, gX = hi ? go : gm, rX = hi ? ro : rm;
        float bY = hi ? bm : bo, gY = hi ? gm : go, rY = hi ? rm : ro;

        v8f cX = run16(bX, gX, rX);
        v8f cY = run16(bY, gY, rY);

        // D layout: VGPR ch, lanes 0-15 = channel ch for 16 consecutive pixels.
        // Redistribute Y into upper lanes so all 32 lanes store coalesced.
        #pragma unroll
        for (int ch = 0; ch < 3; ++ch) {
            float vx  = fminf(fmaxf(cX[ch], 0.0f), 1.0f);
            float vy  = fminf(fmaxf(cY[ch], 0.0f), 1.0f);
            float vys = __shfl_xor(vy, 16, 32);
            float v   = hi ? vys : vx;
            float* op = (ch == 0) ? ob : (ch == 1) ? og : orp;
            __builtin_nontemporal_store(v, op + idx);
        }
    }
}

extern "C" void kernel_launch(void* const* d_in, const int* in_sizes, int n_in,
                              void* d_out, int out_size, void* d_ws, size_t ws_size,
                              hipStream_t stream) {
    (void)in_sizes; (void)n_in; (void)out_size; (void)d_ws; (void)ws_size;
    const float* img    = (const float*)d_in[0];
    const float* params = (const float*)d_in[1];
    float*       out    = (float*)d_out;
    hipLaunchKernelGGL(WbQuadratic_wmma_kernel,
                       dim3(GRID), dim3(BLOCK), 0, stream,
                       img, params, out);
}